// PreciseRoIPooling_71451075936449
// MI455X (gfx1250) — compile-verified
//
#include <hip/hip_runtime.h>

// PrRoIPool (Precise RoI Pooling) for MI455X / gfx1250.
// Fused per-ROI GEMM on the fp32 WMMA pipe: V_WMMA_F32_16X16X4_F32.
//   out[c,(q,p)] = (1/area) * sum_{h,w} F[b,c,h,w] * Wy[q,h] * Wx[p,w]
// M = 256 channels (16 tiles), N = 49->64 bins (4 tiles), K iterated as
// (h outer, w in chunks of 4) with nx padded to a multiple of 4 so the hot
// loop has no division and no predicated loads.

typedef __attribute__((ext_vector_type(2))) float v2f;
typedef __attribute__((ext_vector_type(8))) float v8f;

#define POOLED   7
#define SCALEF   0.0625f
#define CDIM     256
#define HDIM     100
#define WDIM     100
#define MAXS     48     // max window samples per axis (actual bound: 33; 48 floats = 192B rows -> 8B-aligned pairs)

__device__ __forceinline__ float hat_cdf(float t) {
    // G(t) = integral_{-inf}^{t} max(0, 1-|s|) ds
    float u = fminf(1.0f, fmaxf(-1.0f, t));
    float a = u + 1.0f;
    float b = 1.0f - u;
    return (u < 0.0f) ? (0.5f * a * a) : (1.0f - 0.5f * b * b);
}

__global__ __launch_bounds__(256)
void prroi_wmma_kernel(const float* __restrict__ feat,
                       const float* __restrict__ rois,
                       float* __restrict__ out) {
    __shared__ float sWx[POOLED][MAXS];
    __shared__ float sWy[POOLED][MAXS];

    const int n    = blockIdx.x;
    const int tid  = threadIdx.x;
    const int lane = tid & 31;
    const int wave = tid >> 5;          // 0..7

    // ---- ROI parameters (uniform across block; recomputed per thread) ----
    const float* r = rois + (long)n * 5;
    const int   b  = (int)r[0];
    const float x1 = r[1] * SCALEF, y1 = r[2] * SCALEF;
    const float x2 = r[3] * SCALEF, y2 = r[4] * SCALEF;
    const float bw = (x2 - x1) * (1.0f / POOLED);
    const float bh = (y2 - y1) * (1.0f / POOLED);

    // samples with nonzero hat weight: i in (x1-1, x2+1), clipped to the map
    int x_lo = max(0, (int)ceilf(x1 - 1.0f));
    int x_hi = min(WDIM - 1, (int)floorf(x2 + 1.0f));
    int y_lo = max(0, (int)ceilf(y1 - 1.0f));
    int y_hi = min(HDIM - 1, (int)floorf(y2 + 1.0f));
    const int nx  = min(MAXS, max(1, x_hi - x_lo + 1));
    const int ny  = min(MAXS, max(1, y_hi - y_lo + 1));
    const int nx4 = (nx + 3) & ~3;      // w padded to K-chunk multiple

    // Input distribution guarantees a non-degenerate window (ROI sides are
    // >= 2 feature units and start inside the map); tell the compiler so it
    // drops the loop-bypass path (kills the acc phi-copy storm).
    __builtin_assume(nx >= 1 && nx <= MAXS);
    __builtin_assume(ny >= 1 && ny <= MAXS);
    __builtin_assume(nx4 >= 4 && (nx4 & 3) == 0);

    const float area     = bw * bh;
    const float inv_area = (area > 0.0f) ? (1.0f / area) : 0.0f;

    // ---- stage hat-integral weights into LDS (zero-padded rows) ----
    for (int idx = tid; idx < POOLED * MAXS; idx += 256) {
        int p = idx / MAXS;
        int i = idx - p * MAXS;
        float e0x = x1 + bw * (float)p, e1x = e0x + bw;
        float fx  = (float)(x_lo + i);
        sWx[p][i] = (i < nx) ? (hat_cdf(e1x - fx) - hat_cdf(e0x - fx)) : 0.0f;
        float e0y = y1 + bh * (float)p, e1y = e0y + bh;
        float fy  = (float)(y_lo + i);
        sWy[p][i] = (i < ny) ? (hat_cdf(e1y - fy) - hat_cdf(e0y - fy)) : 0.0f;
    }
    __syncthreads();

    // ---- accumulators: wave owns channel tiles {wave, wave+8}, 4 N-tiles ----
    v8f acc[2][4];
    #pragma unroll
    for (int t = 0; t < 2; ++t)
        #pragma unroll
        for (int nt = 0; nt < 4; ++nt)
            acc[t][nt] = {};

    const int m  = lane & 15;                 // A row (channel-in-tile) / B col
    const int kf = (lane < 16) ? 0 : 2;       // K pair held by this half-wave

    // per-lane (q,p) for each N-tile column, padded cols masked to zero
    int   cq[4], cp[4];
    float cmask[4];
    #pragma unroll
    for (int nt = 0; nt < 4; ++nt) {
        int col   = nt * 16 + m;
        int q     = col / POOLED;             // constant divisor -> mul/shift
        int p     = col - q * POOLED;
        cmask[nt] = (col < POOLED * POOLED) ? 1.0f : 0.0f;
        cq[nt]    = min(q, POOLED - 1);
        cp[nt]    = min(p, POOLED - 1);
    }

    const long fbase = (long)b * CDIM * (HDIM * (long)WDIM);
    const int  c0 = wave * 16 + m;            // channel for A tile 0
    const int  c1 = (wave + 8) * 16 + m;      // channel for A tile 1
    const float* fptr0 = feat + fbase + (long)c0 * (HDIM * WDIM);
    const float* fptr1 = feat + fbase + (long)c1 * (HDIM * WDIM);

    for (int h = 0; h < ny; ++h) {
        // Wy factor for this row, per N-tile (masked for padded columns)
        float wy[4];
        #pragma unroll
        for (int nt = 0; nt < 4; ++nt)
            wy[nt] = cmask[nt] * sWy[cq[nt]][h];

        const int row_off = (y_lo + h) * WDIM + x_lo;

        for (int wc = 0; wc < nx4; wc += 4) {
            // A fragments (16x4 fp32: lanes 0-15 hold K{0,1}, lanes 16-31 K{2,3}).
            // Padded w-slots have B==0, so clamp offsets (stay in-bounds) and
            // load unconditionally -- no EXEC-masked branches.
            const int wa = min(wc + kf,     nx - 1);
            const int wb = min(wc + kf + 1, nx - 1);
            v2f a0, a1;
            a0.x = fptr0[row_off + wa];
            a0.y = fptr0[row_off + wb];
            a1.x = fptr1[row_off + wa];
            a1.y = fptr1[row_off + wb];

            #pragma unroll
            for (int nt = 0; nt < 4; ++nt) {
                // B fragment (4x16 fp32, mirror of A: lane -> N, half-wave -> K pair)
                // Wx pair is 8B-aligned in LDS (wc+kf even, 192B rows) -> ds_load_b64
                v2f wx2 = *(const v2f*)&sWx[cp[nt]][wc + kf];
                v2f bb;
                bb.x = wy[nt] * wx2.x;
                bb.y = wy[nt] * wx2.y;
                acc[0][nt] = __builtin_amdgcn_wmma_f32_16x16x4_f32(
                    false, a0, false, bb, (short)0, acc[0][nt], false, false);
                acc[1][nt] = __builtin_amdgcn_wmma_f32_16x16x4_f32(
                    false, a1, false, bb, (short)0, acc[1][nt], false, false);
            }
        }
    }

    // ---- epilogue: D tile layout (VGPR r: rows r / r+8; lane -> col) ----
    const int row_add = (lane < 16) ? 0 : 8;
    #pragma unroll
    for (int t = 0; t < 2; ++t) {
        const int c_tile = (wave + t * 8) * 16;
        #pragma unroll
        for (int nt = 0; nt < 4; ++nt) {
            const int col = nt * 16 + m;
            if (col < POOLED * POOLED) {
                const int q = col / POOLED, p = col - q * POOLED;
                #pragma unroll
                for (int rr = 0; rr < 8; ++rr) {
                    const int c = c_tile + rr + row_add;
                    out[(((long)n * CDIM + c) * POOLED + q) * POOLED + p] =
                        acc[t][nt][rr] * inv_area;
                }
            }
        }
    }
}

extern "C" void kernel_launch(void* const* d_in, const int* in_sizes, int n_in,
                              void* d_out, int out_size, void* d_ws, size_t ws_size,
                              hipStream_t stream) {
    const float* feat = (const float*)d_in[0];   // [B=4, C=256, H=100, W=100] fp32
    const float* rois = (const float*)d_in[1];   // [N, 5] fp32
    float* out = (float*)d_out;                  // [N, 256, 7, 7] fp32
    const int N = in_sizes[1] / 5;               // 512
    prroi_wmma_kernel<<<dim3(N), dim3(256), 0, stream>>>(feat, rois, out);
}